// BoundaryLoss_20358144983670
// MI455X (gfx1250) — compile-verified
//
#include <hip/hip_runtime.h>
#include <stdint.h>

typedef __attribute__((ext_vector_type(2))) float v2f;
typedef __attribute__((ext_vector_type(8))) float v8f;

#define DDEPTH 96
#define HROWS  256
#define WORDS_PER_ROW 8      // 256 voxels / 32
#define WORDS_PER_SLICE 2048 // 256 rows * 8 words

// ---------------- Stage A: bit-pack targets (1 bit / voxel) ----------------
__global__ void bl_pack_bits(const float* __restrict__ t, uint32_t* __restrict__ bits, int N) {
    int i = blockIdx.x * blockDim.x + threadIdx.x;
    bool b = (i < N) ? (t[i] > 0.5f) : false;
    uint32_t m = __builtin_amdgcn_ballot_w32(b);
    if ((threadIdx.x & 31) == 0 && i < N) bits[i >> 5] = m;
}

// 1,2,3-fold erosion along W of a 32-voxel word (4-bit halos; OOB = foreground = 1)
__device__ __forceinline__ void bl_erodeW3(uint32_t p, uint32_t c, uint32_t n, uint32_t o[3]) {
    unsigned long long e = (unsigned long long)(p >> 28)
                         | ((unsigned long long)c << 4)
                         | ((unsigned long long)(n & 0xFu) << 36)
                         | 0xFFFFFF0000000000ULL;
#pragma unroll
    for (int k = 0; k < 3; ++k) {
        e = e & (e << 1) & (e >> 1);
        o[k] = (uint32_t)(e >> 4);
    }
}

// ---------------- Stage B: W-direction erosions (pos mask and neg mask) ----
__global__ void bl_erodeW(const uint32_t* __restrict__ bits, uint32_t* __restrict__ out, int NW) {
    int wi = blockIdx.x * blockDim.x + threadIdx.x;
    if (wi >= NW) return;
    int ww = wi & (WORDS_PER_ROW - 1);
    uint32_t c = bits[wi];
    bool hasP = (ww > 0), hasN = (ww < WORDS_PER_ROW - 1);
    uint32_t p = hasP ? bits[wi - 1] : 0u;
    uint32_t n = hasN ? bits[wi + 1] : 0u;
    uint32_t pos[3], neg[3];
    bl_erodeW3(hasP ? p : 0xFFFFFFFFu, c, hasN ? n : 0xFFFFFFFFu, pos);
    bl_erodeW3(hasP ? ~p : 0xFFFFFFFFu, ~c, hasN ? ~n : 0xFFFFFFFFu, neg);
#pragma unroll
    for (int k = 0; k < 3; ++k) {
        out[(size_t)k * NW + wi]       = pos[k];
        out[(size_t)(k + 3) * NW + wi] = neg[k];
    }
}

// ---------------- Stage C: H-direction erosion (AND of ±k rows) ------------
__global__ void bl_erodeH(const uint32_t* __restrict__ in, uint32_t* __restrict__ out, int NW) {
    int wi = blockIdx.x * blockDim.x + threadIdx.x;
    if (wi >= NW) return;
    int h = (wi >> 3) & (HROWS - 1);
#pragma unroll
    for (int plane = 0; plane < 6; ++plane) {
        int kk = (plane % 3) + 1;
        const uint32_t* pw = in + (size_t)plane * NW;
        uint32_t a = pw[wi];
#pragma unroll
        for (int j = 1; j <= 3; ++j) {
            if (j <= kk) {
                if (h - j >= 0)    a &= pw[wi - j * WORDS_PER_ROW];
                if (h + j < HROWS) a &= pw[wi + j * WORDS_PER_ROW];
            }
        }
        out[(size_t)plane * NW + wi] = a;
    }
}

// -------- Stage D: D-direction erosion + sigmoid*sd*valid + reduction ------
__global__ void bl_main(const float* __restrict__ logits, const float* __restrict__ valid,
                        const uint32_t* __restrict__ bits, const uint32_t* __restrict__ hC,
                        float* __restrict__ acc, int NW) {
    const int wave = __builtin_amdgcn_readfirstlane((int)(threadIdx.x >> 5));
    const int lane = threadIdx.x & 31;
    const int wpb  = blockDim.x >> 5;
    const int stride = gridDim.x * wpb;

    float accT = 0.0f, accV = 0.0f;

    for (int wi = blockIdx.x * wpb + wave; wi < NW; wi += stride) {
        if (wi + stride < NW)
            __builtin_prefetch(&logits[(size_t)(wi + stride) * 32], 0, 1);

        int d = (wi / WORDS_PER_SLICE) % DDEPTH;
        uint32_t tw = bits[wi];

        uint32_t ep[3], en[3];
#pragma unroll
        for (int k = 0; k < 3; ++k) {
            ep[k] = hC[(size_t)k * NW + wi];
            en[k] = hC[(size_t)(k + 3) * NW + wi];
#pragma unroll
            for (int j = 1; j <= 3; ++j) {
                if (j <= k + 1) {
                    if (d - j >= 0) {
                        ep[k] &= hC[(size_t)k * NW + wi - j * WORDS_PER_SLICE];
                        en[k] &= hC[(size_t)(k + 3) * NW + wi - j * WORDS_PER_SLICE];
                    }
                    if (d + j < DDEPTH) {
                        ep[k] &= hC[(size_t)k * NW + wi + j * WORDS_PER_SLICE];
                        en[k] &= hC[(size_t)(k + 3) * NW + wi + j * WORDS_PER_SLICE];
                    }
                }
            }
        }

        float l = logits[(size_t)wi * 32 + lane];
        float v = valid[(size_t)wi * 32 + lane];
        float prob = 1.0f / (1.0f + __expf(-l));
        int tb = (int)((tw >> lane) & 1u);
        int ps = (int)((ep[0] >> lane) & 1u) + (int)((ep[1] >> lane) & 1u) + (int)((ep[2] >> lane) & 1u);
        int ns = (int)((en[0] >> lane) & 1u) + (int)((en[1] >> lane) & 1u) + (int)((en[2] >> lane) & 1u);
        float sd = tb ? -(float)(1 + ps) : (float)(1 + ns);
        accT += prob * sd * v;
        accV += v;
    }

    // Wave-level reduction with v_wmma_f32_16x16x4_f32: B = ones, A col0 = partials.
    // D[m][n] = acc[m] + acc[m+16]; summing the 8 C regs gives half-sums, lane^16 fixes it.
    v2f b; b.x = 1.0f; b.y = 1.0f;
    v8f c0 = {};
    v2f aT; aT.x = accT; aT.y = 0.0f;
    v8f dT = __builtin_amdgcn_wmma_f32_16x16x4_f32(false, aT, false, b, (short)0, c0, false, false);
    v2f aV; aV.x = accV; aV.y = 0.0f;
    v8f dV = __builtin_amdgcn_wmma_f32_16x16x4_f32(false, aV, false, b, (short)0, c0, false, false);

    float sT = 0.0f, sV = 0.0f;
#pragma unroll
    for (int r = 0; r < 8; ++r) { sT += dT[r]; sV += dV[r]; }
    sT += __shfl_xor(sT, 16, 32);
    sV += __shfl_xor(sV, 16, 32);

    if (lane == 0) {
        atomicAdd(&acc[0], sT);
        atomicAdd(&acc[1], sV);
    }
}

__global__ void bl_init(float* acc) { acc[0] = 0.0f; acc[1] = 0.0f; }
__global__ void bl_finalize(const float* __restrict__ acc, float* __restrict__ out) {
    out[0] = acc[0] / fmaxf(acc[1], 1.0f);
}

extern "C" void kernel_launch(void* const* d_in, const int* in_sizes, int n_in,
                              void* d_out, int out_size, void* d_ws, size_t ws_size,
                              hipStream_t stream) {
    const float* logits  = (const float*)d_in[0];
    const float* targets = (const float*)d_in[1];
    const float* valid   = (const float*)d_in[2];
    float* out = (float*)d_out;

    int N  = in_sizes[0];          // 2*1*96*256*256 = 12,582,912
    int NW = N >> 5;               // packed 32-voxel words

    char* ws = (char*)d_ws;
    float*    acc  = (float*)ws;                                   // 2 floats
    uint32_t* bits = (uint32_t*)(ws + 256);                        // NW words (1.5 MB)
    uint32_t* stB  = (uint32_t*)(ws + 256 + (size_t)NW * 4);       // 6 planes W-eroded
    uint32_t* stC  = (uint32_t*)(ws + 256 + (size_t)NW * 4 * 7);   // 6 planes H-eroded

    bl_init<<<1, 1, 0, stream>>>(acc);
    bl_pack_bits<<<(N + 255) / 256, 256, 0, stream>>>(targets, bits, N);
    bl_erodeW<<<(NW + 255) / 256, 256, 0, stream>>>(bits, stB, NW);
    bl_erodeH<<<(NW + 255) / 256, 256, 0, stream>>>(stB, stC, NW);

    int blocks = 2048;
    if (blocks > (NW + 7) / 8) blocks = (NW + 7) / 8;
    bl_main<<<blocks, 256, 0, stream>>>(logits, valid, bits, stC, acc, NW);

    bl_finalize<<<1, 1, 0, stream>>>(acc, out);
}